// IIR_SSRT_RenderLayer_35734127903424
// MI455X (gfx1250) — compile-verified
//
#include <hip/hip_runtime.h>
#include <math.h>

namespace {

constexpr int kH   = 480;
constexpr int kW   = 640;
constexpr int kSPP = 16;
constexpr int kHW  = kH * kW;
constexpr int BLK  = 256;
constexpr int kBufFloats = 7 * BLK;   // per staging buffer: 4*BLK (wo+pdf) + 3*BLK (light)

static_assert(kHW % BLK == 0, "grid must tile exactly");

constexpr float kInvPi = 0.31830988618379067154f;
constexpr float kPi    = 3.14159265358979323846f;

// tan(85deg/2) = tan(42.5deg), precomputed to double precision.
constexpr double kXRd = 0.9163311743;
constexpr double kYRd = 0.75 * kXRd;           // (H/W) * xR
constexpr float  kSX  = (float)(2.0 * kXRd / (double)(kW - 1));
constexpr float  kSY  = (float)(2.0 * kYRd / (double)(kH - 1));
constexpr float  kX0  = (float)(-kXRd);
constexpr float  kY0  = (float)(kYRd);

__device__ __forceinline__ float rcpf_(float x)  { return __builtin_amdgcn_rcpf(x); }
__device__ __forceinline__ float rsqf_(float x)  { return __builtin_amdgcn_rsqf(x); }
__device__ __forceinline__ float sqrtf_(float x) { return __builtin_amdgcn_sqrtf(x); }

// CDNA5 async global->LDS copy (ASYNCcnt-tracked, no VGPR data path).
// GVS mode: mem_addr = SGPR64 + VGPR32_offset ; LDS dest addr from VGPR.
__device__ __forceinline__ void async_ld_b32(unsigned lds_addr,
                                             const float* sbase,
                                             int voff_bytes) {
  asm volatile("global_load_async_to_lds_b32 %0, %1, %2"
               :: "v"(lds_addr), "v"(voff_bytes), "s"(sbase)
               : "memory");
}

__device__ __forceinline__ void wait_async_le7() {
  asm volatile("s_wait_asynccnt 0x7" ::: "memory");
}
__device__ __forceinline__ void wait_async_le0() {
  asm volatile("s_wait_asynccnt 0x0" ::: "memory");
}
// Each iteration issues >=2 DS reads; LDS completes in order, so DScnt<=2
// guarantees every *older* iteration's DS reads have fully landed.
__device__ __forceinline__ void wait_ds_le2() {
  asm volatile("s_wait_dscnt 0x2" ::: "memory");
}

} // namespace

__global__ __launch_bounds__(BLK)
void render_ggx_kernel(const float* __restrict__ albedo,
                       const float* __restrict__ rough,
                       const float* __restrict__ metal,
                       const float* __restrict__ normal,
                       const float* __restrict__ wo_emitter,
                       const float* __restrict__ light,
                       const float* __restrict__ pdf_emitter,
                       float* __restrict__ out)
{
  // Triple-buffered staging. Per lane: float4 {wo.xyz, pdf} + float3 light.
  __shared__ __align__(16) float smem[3][kBufFloats];

  const int lane = threadIdx.x;
  const int pix  = blockIdx.x * BLK + lane;

  // LDS byte offset of smem (generic-pointer low 32 bits == LDS offset).
  const unsigned lds0 = (unsigned)(unsigned long long)(void*)(&smem[0][0]);

  auto stage = [&](int s, int buf) {
    const unsigned base  = lds0 + (unsigned)(buf * kBufFloats * 4);
    const int      woOff = ((s * 3) * kHW + pix) * 4;
    // Interleaved region A: lane-local float4 -> one ds_load_b128 later.
    async_ld_b32(base + (lane * 4 + 0) * 4, wo_emitter, woOff);
    async_ld_b32(base + (lane * 4 + 1) * 4, wo_emitter, woOff + kHW * 4);
    async_ld_b32(base + (lane * 4 + 2) * 4, wo_emitter, woOff + 2 * kHW * 4);
    async_ld_b32(base + (lane * 4 + 3) * 4, pdf_emitter, (s * kHW + pix) * 4);
    // Region B: lane-local float3 (light is HWC-interleaved in memory).
    const int liOff = (s * kHW + pix) * 12;
    async_ld_b32(base + (4 * BLK + 3 * lane + 0) * 4, light, liOff + 0);
    async_ld_b32(base + (4 * BLK + 3 * lane + 1) * 4, light, liOff + 4);
    async_ld_b32(base + (4 * BLK + 3 * lane + 2) * 4, light, liOff + 8);
  };

  // Kick off the pipeline: samples 0 and 1 in flight.
  stage(0, 0);
  stage(1, 1);

  // ---- per-pixel constant work (overlaps with in-flight async loads) ----
  const int h = pix / kW;
  const int w = pix - h * kW;

  // V_DIR: p = (x, y, -1), v = normalize(-p). Pure f32, constants folded.
  const float vx0 = -fmaf((float)w,  kSX, kX0);
  const float vy0 = -fmaf((float)h, -kSY, kY0);
  const float vri = rsqf_(fmaxf(vx0 * vx0 + vy0 * vy0 + 1.0f, 1e-12f));
  const float Vx = vx0 * vri, Vy = vy0 * vri, Vz = vri;

  // Pixel-constant inputs (planar, fully coalesced).
  const float nx = normal[pix];
  const float ny = normal[kHW + pix];
  const float nz = normal[2 * kHW + pix];
  const float A0 = albedo[pix];
  const float A1 = albedo[kHW + pix];
  const float A2 = albedo[2 * kHW + pix];
  const float rr = rough[pix];
  const float mm = metal[pix];

  // Tangent frame (branchless ONB).
  const float nlen = sqrtf_(nx * nx + ny * ny + nz * nz);
  const float ninv = rcpf_(fmaxf(nlen, 1e-6f));
  const float zx = nx * ninv, zy = ny * ninv, zz = nz * ninv;
  const float sgn = (zz >= 0.0f) ? 1.0f : -1.0f;
  const float fa  = -rcpf_(sgn + zz);
  const float fb  = zx * zy * fa;
  const float cxx = 1.0f + sgn * zx * zx * fa, cxy = sgn * fb, cxz = -sgn * zx;
  const float cyx = fb, cyy = sgn + zy * zy * fa, cyz = -zy;

  // wi = frame * V_DIR, z clamped + renormalized.
  const float wi0x = cxx * Vx + cxy * Vy + cxz * Vz;
  const float wi0y = cyx * Vx + cyy * Vy + cyz * Vz;
  const float wi0z = zx * Vx + zy * Vy + zz * Vz;
  const float wiz_abs = fabsf(wi0z);
  const float wmask   = (wiz_abs < 1e-6f) ? 0.0f : 1.0f;
  const float wzc  = fmaxf(wiz_abs, 0.001f);
  const float wlen = sqrtf_(wi0x * wi0x + wi0y * wi0y + wzc * wzc);
  const float winv = rcpf_(fmaxf(wlen, 1e-6f));
  const float wix = wi0x * winv, wiy = wi0y * winv, wizn = wzc * winv;
  // wizn = ndv: provably > 0 (wzc >= 1e-3, winv > 0), so the GGX mask
  // (ndl>0 && ndv>0) reduces to (ndl > 0).

  // BRDF constants.
  const float one_m = 1.0f - mm;
  const float D0 = A0 * one_m * kInvPi;
  const float D1 = A1 * one_m * kInvPi;
  const float D2 = A2 * one_m * kInvPi;
  const float F00 = 0.04f * one_m + A0 * mm;
  const float F01 = 0.04f * one_m + A1 * mm;
  const float F02 = 0.04f * one_m + A2 * mm;
  const float alpha = fmaxf(rr * rr, 1e-4f);
  const float a2    = alpha * alpha;
  const float kk    = 0.5f * alpha;
  const float omk   = 1.0f - kk;
  const float gv4pi = 4.0f * kPi * (wizn * omk + kk); // const part of G denom

  // Accumulators: shading, masked-lt (diffuse), Fresnel-split spec (A,B).
  float sh0 = 0.f, sh1 = 0.f, sh2 = 0.f;
  float md0 = 0.f, md1 = 0.f, md2 = 0.f;
  float sA0 = 0.f, sA1 = 0.f, sA2 = 0.f;
  float sB0 = 0.f, sB1 = 0.f, sB2 = 0.f;

#pragma unroll
  for (int s = 0; s < kSPP; ++s) {
    const int buf = s % 3;
    // Stage `s` was issued 2 iterations ago; keep the next one in flight.
    if (s < kSPP - 1) wait_async_le7();
    else              wait_async_le0();

    // Wide LDS reads: one b128 (wo.xyz+pdf) + one b96 (light.rgb).
    const float4 wp = *reinterpret_cast<const float4*>(&smem[buf][lane * 4]);
    const float wex = wp.x, wey = wp.y, wez = wp.z, pdf = wp.w;
    const float* lb = &smem[buf][4 * BLK + 3 * lane];
    const float L0 = lb[0], L1 = lb[1], L2 = lb[2];

    // Re-stage buffer (s+2)%3 (last read at iteration s-1): only *older*
    // DS reads must have landed -> near-free DScnt<=2 wait (in-order LDS).
    if (s + 2 < kSPP) {
      wait_ds_le2();
      stage(s + 2, (s + 2) % 3);
    }

    // wo = frame * wo_emitter, z := |z|.
    const float tx = cxx * wex + cxy * wey + cxz * wez;
    const float ty = cyx * wex + cyy * wey + cyz * wez;
    const float tz = fabsf(zx * wex + zy * wey + zz * wez);

    // Half vector.
    const float hx0 = wix + tx, hy0 = wiy + ty, hz0 = wizn + tz;
    const float hinv =
        rcpf_(fmaxf(sqrtf_(hx0 * hx0 + hy0 * hy0 + hz0 * hz0), 1e-6f));
    const float hx = hx0 * hinv, hy = hy0 * hinv, hz = hz0 * hinv;

    const float ndl = tz;                       // max(|z|, 0)
    const float ndh = fmaxf(hz, 0.0f);
    const float vdh = fmaxf(wix * hx + wiy * hy + wizn * hz, 0.0f);

    const float t  = 1.0f - vdh;
    const float t2 = t * t;
    const float t5 = t2 * t2 * t;

    // Fused D/(4 pi-denominators): G = a2 / (4*pi*dd^2*gv*gl).
    const float dd = ndh * ndh * (a2 - 1.0f) + 1.0f;
    const float G  = a2 * rcpf_(gv4pi * dd * dd * (ndl * omk + kk));
    const float mk = (ndl > 0.0f) ? 1.0f : 0.0f;

    const float sc = ndl * rcpf_(fmaxf(pdf, 0.001f));  // light scale

    const float lt0 = L0 * sc, lt1 = L1 * sc, lt2 = L2 * sc;
    sh0 += lt0; sh1 += lt1; sh2 += lt2;

    const float m0 = mk * lt0, m1 = mk * lt1, m2 = mk * lt2;
    md0 += m0; md1 += m1; md2 += m2;

    const float g0 = G * m0, g1 = G * m1, g2 = G * m2;
    sA0 += g0; sA1 += g1; sA2 += g2;
    sB0 += t5 * g0; sB1 += t5 * g1; sB2 += t5 * g2;
  }

  const float inv_spp = 1.0f / (float)kSPP;

  // colorDiffuse (3 planes)
  out[0 * kHW + pix] = D0 * md0 * inv_spp;
  out[1 * kHW + pix] = D1 * md1 * inv_spp;
  out[2 * kHW + pix] = D2 * md2 * inv_spp;
  // colorSpec (3 planes): F = F0 + (1-F0) t^5  ->  F0*A + (1-F0)*B
  out[3 * kHW + pix] = (F00 * sA0 + (1.0f - F00) * sB0) * inv_spp;
  out[4 * kHW + pix] = (F01 * sA1 + (1.0f - F01) * sB1) * inv_spp;
  out[5 * kHW + pix] = (F02 * sA2 + (1.0f - F02) * sB2) * inv_spp;
  // wi_mask (1 plane)
  out[6 * kHW + pix] = wmask;
  // shading (3 planes)
  out[7 * kHW + pix] = sh0 * inv_spp;
  out[8 * kHW + pix] = sh1 * inv_spp;
  out[9 * kHW + pix] = sh2 * inv_spp;
}

extern "C" void kernel_launch(void* const* d_in, const int* in_sizes, int n_in,
                              void* d_out, int out_size, void* d_ws, size_t ws_size,
                              hipStream_t stream) {
  (void)in_sizes; (void)n_in; (void)out_size; (void)d_ws; (void)ws_size;
  const float* albedo     = (const float*)d_in[0];
  const float* rough      = (const float*)d_in[1];
  const float* metal      = (const float*)d_in[2];
  const float* normal     = (const float*)d_in[3];
  /* d_in[4] = vpos — unused by the reference */
  const float* wo_emitter = (const float*)d_in[5];
  const float* light      = (const float*)d_in[6];
  const float* pdf        = (const float*)d_in[7];
  float* out = (float*)d_out;

  hipLaunchKernelGGL(render_ggx_kernel,
                     dim3(kHW / BLK), dim3(BLK), 0, stream,
                     albedo, rough, metal, normal, wo_emitter, light, pdf, out);
}